// FlexibleClassifier_429496730317
// MI455X (gfx1250) — compile-verified
//
#include <hip/hip_runtime.h>

typedef float v2f __attribute__((ext_vector_type(2)));
typedef float v8f __attribute__((ext_vector_type(8)));

constexpr int NNODES = 8192;
constexpr int DFEAT  = 256;
constexpr int NCLS   = 4;
constexpr int MSPLIT = 8;                    // m-dimension split for occupancy
constexpr int MSEG   = NNODES / MSPLIT;      // 1024 m per block
constexpr int CHUNK  = 512;                  // g rows staged in LDS per pass
constexpr int WAVES  = 4;                    // waves per block (wave32)
constexpr int ROWS_PER_BLOCK = WAVES * 16;   // 64 output rows per block

// ---------------------------------------------------------------------------
// y = X @ W + b ; g0 = softmax(y)  (row-local softmax, C = 4)
// ---------------------------------------------------------------------------
__global__ __launch_bounds__(256) void crf_init_kernel(
    const float* __restrict__ X, const float* __restrict__ W,
    const float* __restrict__ bvec, float* __restrict__ y,
    float* __restrict__ g)
{
  __shared__ float Ws[DFEAT * NCLS];
  __shared__ float ys[64 * NCLS];
  const int tid = threadIdx.x;
  for (int i = tid; i < DFEAT * NCLS; i += 256) Ws[i] = W[i];
  __syncthreads();

  const int lr  = tid >> 2;                  // local row 0..63
  const int c   = tid & 3;                   // class
  const int row = blockIdx.x * 64 + lr;
  const float* xr = X + (size_t)row * DFEAT;
  float acc = 0.f;
  #pragma unroll 4
  for (int d = 0; d < DFEAT; d += 4) {
    float4 xv = *(const float4*)(xr + d);
    acc = fmaf(xv.x, Ws[(d + 0) * NCLS + c], acc);
    acc = fmaf(xv.y, Ws[(d + 1) * NCLS + c], acc);
    acc = fmaf(xv.z, Ws[(d + 2) * NCLS + c], acc);
    acc = fmaf(xv.w, Ws[(d + 3) * NCLS + c], acc);
  }
  acc += bvec[c];
  y[(size_t)row * NCLS + c] = acc;
  ys[lr * NCLS + c] = acc;
  __syncthreads();

  const float v0 = ys[lr * 4 + 0], v1 = ys[lr * 4 + 1];
  const float v2 = ys[lr * 4 + 2], v3 = ys[lr * 4 + 3];
  const float mx = fmaxf(fmaxf(v0, v1), fmaxf(v2, v3));
  const float es = __expf(v0 - mx) + __expf(v1 - mx) +
                   __expf(v2 - mx) + __expf(v3 - mx);
  g[(size_t)row * NCLS + c] = __expf(acc - mx) / es;
}

// ---------------------------------------------------------------------------
// partial[mseg] = sw0 * f0[tile, mseg] @ g  +  sw1 * f1[tile, mseg] @ g
// One 16-row tile per wave; WMMA f32 16x16x4; f streamed non-temporally.
// ---------------------------------------------------------------------------
__global__ __launch_bounds__(32 * WAVES) void crf_msg_kernel(
    const float* __restrict__ f, const float* __restrict__ gin,
    const float* __restrict__ sw, float* __restrict__ partial)
{
  __shared__ float gsm[CHUNK * NCLS];          // 8 KB staged g chunk
  __shared__ float msg_s[WAVES * 16 * NCLS];   // per-wave tile staging

  const int tid  = threadIdx.x;
  const int wave = tid >> 5;
  const int lane = tid & 31;
  const int kh   = lane >> 4;                  // 0 -> K{0,1}, 1 -> K{2,3}
  const int r    = lane & 15;                  // A row / D column

  const int rowTile = blockIdx.x * ROWS_PER_BLOCK + wave * 16;
  const int mseg    = blockIdx.y;
  const int mbase   = mseg * MSEG;

  const size_t frow = (size_t)(rowTile + r) * NNODES;
  const float* f0 = f + frow;                               // kernel 0
  const float* f1 = f + (size_t)NNODES * NNODES + frow;     // kernel 1

  const int bn = r & 3;  // replicate class columns across the 16 N-lanes

  v8f s0 = {};
  v8f s1 = {};

  for (int cb = mbase; cb < mbase + MSEG; cb += CHUNK) {
    __syncthreads();
    {  // cooperative stage of g[cb .. cb+CHUNK) into LDS (hot in L2)
      const float4* src = (const float4*)(gin + (size_t)cb * NCLS);
      float4* dst = (float4*)gsm;
      #pragma unroll
      for (int i = 0; i < (CHUNK * NCLS / 4) / (32 * WAVES); ++i)
        dst[tid + i * 32 * WAVES] = src[tid + i * 32 * WAVES];
    }
    __syncthreads();

    const float* fp0 = f0 + cb + 2 * kh;
    const float* fp1 = f1 + cb + 2 * kh;
    #pragma unroll 8
    for (int mm = 0; mm < CHUNK; mm += 4) {
      // A tile (16x4 fp32): lane holds f[row r][m + 2*kh + {0,1}]
      v2f a0 = __builtin_nontemporal_load((const v2f*)(fp0 + mm));
      v2f a1 = __builtin_nontemporal_load((const v2f*)(fp1 + mm));
      // B tile (4x16 fp32): lane n holds g[m + 2*kh + {0,1}][n&3]
      v2f bb;
      bb.x = gsm[(mm + 2 * kh) * NCLS + bn];
      bb.y = gsm[(mm + 2 * kh + 1) * NCLS + bn];
      s0 = __builtin_amdgcn_wmma_f32_16x16x4_f32(
          false, a0, false, bb, (short)0, s0, false, false);
      s1 = __builtin_amdgcn_wmma_f32_16x16x4_f32(
          false, a1, false, bb, (short)0, s1, false, false);
    }
  }

  const float w0 = sw[0], w1 = sw[1];
  if (r < NCLS) {  // D: row = v + 8*kh, col = r (only cols 0..3 meaningful)
    #pragma unroll
    for (int v = 0; v < 8; ++v)
      msg_s[wave * 64 + (v + 8 * kh) * NCLS + r] = w0 * s0[v] + w1 * s1[v];
  }
  __syncthreads();
  if (lane < 16) {
    const int row = rowTile + lane;
    float4 mv;
    mv.x = msg_s[wave * 64 + lane * NCLS + 0];
    mv.y = msg_s[wave * 64 + lane * NCLS + 1];
    mv.z = msg_s[wave * 64 + lane * NCLS + 2];
    mv.w = msg_s[wave * 64 + lane * NCLS + 3];
    *(float4*)(partial + ((size_t)mseg * NNODES + row) * NCLS) = mv;
  }
}

// ---------------------------------------------------------------------------
// msg = sum over m-splits; h = y - (rowsum(msg) - msg); gout = softmax(h)
// ---------------------------------------------------------------------------
__global__ __launch_bounds__(256) void crf_fin_kernel(
    const float* __restrict__ partial, const float* __restrict__ y,
    float* __restrict__ gout)
{
  __shared__ float hs[64 * NCLS];
  const int tid = threadIdx.x;
  const int lr  = tid >> 2;
  const int c   = tid & 3;
  const int row = blockIdx.x * 64 + lr;

  float m = 0.f;
  #pragma unroll
  for (int s = 0; s < MSPLIT; ++s)
    m += partial[((size_t)s * NNODES + row) * NCLS + c];
  hs[lr * 4 + c] = m;
  __syncthreads();
  const float rs = hs[lr * 4 + 0] + hs[lr * 4 + 1] +
                   hs[lr * 4 + 2] + hs[lr * 4 + 3];
  const float h = y[(size_t)row * NCLS + c] - (rs - m);
  __syncthreads();
  hs[lr * 4 + c] = h;
  __syncthreads();
  const float h0 = hs[lr * 4 + 0], h1 = hs[lr * 4 + 1];
  const float h2 = hs[lr * 4 + 2], h3 = hs[lr * 4 + 3];
  const float mx = fmaxf(fmaxf(h0, h1), fmaxf(h2, h3));
  const float es = __expf(h0 - mx) + __expf(h1 - mx) +
                   __expf(h2 - mx) + __expf(h3 - mx);
  gout[(size_t)row * NCLS + c] = __expf(h - mx) / es;
}

// ---------------------------------------------------------------------------
extern "C" void kernel_launch(void* const* d_in, const int* in_sizes, int n_in,
                              void* d_out, int out_size, void* d_ws, size_t ws_size,
                              hipStream_t stream)
{
  (void)in_sizes; (void)n_in; (void)out_size; (void)ws_size;
  const float* X  = (const float*)d_in[0];
  const float* f  = (const float*)d_in[1];
  const float* W  = (const float*)d_in[2];
  const float* b  = (const float*)d_in[3];
  const float* sw = (const float*)d_in[4];
  float* out = (float*)d_out;

  float* ws      = (float*)d_ws;
  float* y       = ws;                                   // 32768 floats
  float* gA      = ws + (size_t)NNODES * NCLS;           // 32768 floats
  float* partial = gA + (size_t)NNODES * NCLS;           // MSPLIT*32768 floats

  const dim3 mgrid(NNODES / ROWS_PER_BLOCK, MSPLIT);     // (128, 8)

  crf_init_kernel<<<NNODES / 64, 256, 0, stream>>>(X, W, b, y, gA);

  // iter 1: g1 = softmax(y - compat(msg(gA)))  -> stage into d_out
  crf_msg_kernel<<<mgrid, 32 * WAVES, 0, stream>>>(f, gA, sw, partial);
  crf_fin_kernel<<<NNODES / 64, 256, 0, stream>>>(partial, y, out);

  // iter 2 -> gA
  crf_msg_kernel<<<mgrid, 32 * WAVES, 0, stream>>>(f, out, sw, partial);
  crf_fin_kernel<<<NNODES / 64, 256, 0, stream>>>(partial, y, gA);

  // iter 3 -> final output (softmax already applied)
  crf_msg_kernel<<<mgrid, 32 * WAVES, 0, stream>>>(f, gA, sw, partial);
  crf_fin_kernel<<<NNODES / 64, 256, 0, stream>>>(partial, y, out);
}